// NativeSparseAttention_51857435131910
// MI455X (gfx1250) — compile-verified
//
#include <hip/hip_runtime.h>

// Problem constants (reference: B=2, H=8, S=2048, D=64)
#define BB 2
#define HH 8
#define SS 2048
#define DD 64
#define NB 32        // number of key blocks
#define BS 64        // block size
#define NSEL 8       // selected blocks
#define SELC 512     // selected columns = NSEL*BS
#define KEEP 256     // min(NUM_SELECTED_TOKENS*NSEL, SELC)
#define WIN 128      // sliding window
#define PSTRIDE 520  // padded LDS row stride (floats)
#define NEGINF (-1e30f)
#define SCALE 0.125f // 1/sqrt(64)

typedef __attribute__((ext_vector_type(2))) float v2f;
typedef __attribute__((ext_vector_type(8))) float v8f;

static __device__ inline v8f vzero8() {
  v8f r;
#pragma unroll
  for (int i = 0; i < 8; ++i) r[i] = 0.f;
  return r;
}

// fp32 WMMA: D(16x16) = A(16x4) * B(4x16) + C   -> v_wmma_f32_16x16x4_f32
static __device__ inline v8f wmma4(v2f a, v2f b, v8f c) {
  return __builtin_amdgcn_wmma_f32_16x16x4_f32(false, a, false, b, (short)0, c,
                                               false, false);
}

// order-preserving float -> uint key (larger key == larger float)
static __device__ inline unsigned okey(float x) {
  unsigned u = __float_as_uint(x);
  return (u & 0x80000000u) ? ~u : (u | 0x80000000u);
}

static __device__ inline int wred_sumi(int x) {
#pragma unroll
  for (int o = 16; o > 0; o >>= 1) x += __shfl_xor(x, o, 32);
  return x;
}
static __device__ inline float wred_sumf(float x) {
#pragma unroll
  for (int o = 16; o > 0; o >>= 1) x += __shfl_xor(x, o, 32);
  return x;
}
static __device__ inline float wred_maxf(float x) {
#pragma unroll
  for (int o = 16; o > 0; o >>= 1) x = fmaxf(x, __shfl_xor(x, o, 32));
  return x;
}

// ---------------- kernel 1: per-(b,h,block) K/V means ----------------
__global__ void nsa_blockmean_kernel(const float* __restrict__ K,
                                     const float* __restrict__ V,
                                     float* __restrict__ KB,
                                     float* __restrict__ VB) {
  const int g = blockIdx.x;   // (b*H + h)*NB + m  (block rows are contiguous)
  const int d = threadIdx.x;  // 0..63
  const float* kp = K + (size_t)g * (BS * DD);
  const float* vp = V + (size_t)g * (BS * DD);
  float ks = 0.f, vs = 0.f;
  for (int t = 0; t < BS; ++t) {
    ks += kp[t * DD + d];
    vs += vp[t * DD + d];
  }
  KB[(size_t)g * DD + d] = ks * (1.f / BS);
  VB[(size_t)g * DD + d] = vs * (1.f / BS);
}

// ---------------- kernel 2: NSA main, one WG per (b,h,qblock) ----------------
__global__ __launch_bounds__(128) void nsa_main_kernel(
    const float* __restrict__ Q, const float* __restrict__ K,
    const float* __restrict__ V, const float* __restrict__ KB,
    const float* __restrict__ VB, float* __restrict__ O) {
  extern __shared__ float sP[];     // 64 rows x PSTRIDE logits/probs
  __shared__ float sScore[NB];      // per-block mean scores
  __shared__ int sSel[NSEL];        // top-8 block ids

  const int wg = blockIdx.x;
  const int qb = wg % NB;
  const int bh = wg / NB;
  const size_t hoff = (size_t)bh * SS * DD;
  const float* Qh = Q + hoff;
  const float* Kh = K + hoff;
  const float* Vh = V + hoff;
  const float* KBh = KB + (size_t)bh * NB * DD;
  const float* VBh = VB + (size_t)bh * NB * DD;
  float* Oh = O + hoff;

  const int tid = threadIdx.x;
  const int wave = tid >> 5;
  const int lane = tid & 31;
  const int half = lane >> 4;   // A/B fragment half: K pair select
  const int ln = lane & 15;     // row (A) / column (B,C) index
  const int qrow0 = qb * BS + wave * 16;  // this wave's 16-query strip
  float* myP = sP + (size_t)wave * 16 * PSTRIDE;

  // Preload Q as 16 fp32 A-fragments covering K(d)=0..63
  v2f qa[16];
#pragma unroll
  for (int t = 0; t < 16; ++t) {
    const float* p = Qh + (size_t)(qrow0 + ln) * DD + 4 * t + 2 * half;
    v2f a; a.x = p[0]; a.y = p[1];
    qa[t] = a;
  }

  v8f oacc[4];  // 16 rows x 64 d output accumulator (4 C tiles)
#pragma unroll
  for (int dt = 0; dt < 4; ++dt) oacc[dt] = vzero8();

  // =============== branch 1: compressed attention ===============
#pragma unroll
  for (int nt = 0; nt < 2; ++nt) {  // 2 column tiles of 16 blocks
    v8f acc = vzero8();
#pragma unroll
    for (int t = 0; t < 16; ++t) {
      const float* bp = KBh + (size_t)(nt * 16 + ln) * DD + 4 * t + 2 * half;
      v2f bf; bf.x = bp[0]; bf.y = bp[1];
      acc = wmma4(qa[t], bf, acc);
    }
#pragma unroll
    for (int r = 0; r < 8; ++r) {
      int m = r + 8 * half;
      int col = nt * 16 + ln;  // block index
      float x = acc[r] * SCALE;
      myP[m * PSTRIDE + col] = (col <= qb) ? x : NEGINF;  // blk_start <= pos
    }
  }
  __syncthreads();

  // block scores = mean over 64 query rows of masked logits
  if (tid < NB) {
    float s = 0.f;
    for (int q2 = 0; q2 < BS; ++q2) s += sP[q2 * PSTRIDE + tid];
    sScore[tid] = s * (1.f / BS);
  }
  __syncthreads();

  // top-8 blocks, JAX tie-break (lowest index first on equal value)
  if (tid == 0) {
    float loc[NB];
    for (int i = 0; i < NB; ++i) loc[i] = sScore[i];
    for (int s = 0; s < NSEL; ++s) {
      int bi = 0;
      float bv = loc[0];
      for (int i = 1; i < NB; ++i)
        if (loc[i] > bv) { bv = loc[i]; bi = i; }
      sSel[s] = bi;
      loc[bi] = -3.4e38f;
    }
  }
  __syncthreads();

  // row softmax over 32 compressed columns (one thread per query row)
  if (tid < BS) {
    float* row = sP + tid * PSTRIDE;
    float mx = -3.4e38f;
    for (int c = 0; c < NB; ++c) mx = fmaxf(mx, row[c]);
    float sum = 0.f;
    for (int c = 0; c < NB; ++c) {
      float p = (mx > -1e29f) ? __expf(row[c] - mx) : 0.f;
      row[c] = p;
      sum += p;
    }
    float rs = (sum > 0.f) ? 1.f / sum : 0.f;
    for (int c = 0; c < NB; ++c) row[c] *= rs;
  }
  __syncthreads();

  // out_cmp: P(16x32) x v_blk(32x64)
#pragma unroll
  for (int dt = 0; dt < 4; ++dt) {
#pragma unroll
    for (int g = 0; g < 8; ++g) {
      int c0 = 4 * g + 2 * half;
      v2f a; a.x = myP[ln * PSTRIDE + c0]; a.y = myP[ln * PSTRIDE + c0 + 1];
      v2f bf;
      bf.x = VBh[(size_t)c0 * DD + dt * 16 + ln];
      bf.y = VBh[(size_t)(c0 + 1) * DD + dt * 16 + ln];
      oacc[dt] = wmma4(a, bf, oacc[dt]);
    }
  }
  __syncthreads();  // logit buffer about to be reused

  // =============== branch 2: selected attention ===============
  for (int sb = 0; sb < NSEL; ++sb) {
    int kb = sSel[sb];
    for (int kt = 0; kt < 4; ++kt) {
      int keyBase = kb * BS + kt * 16;
      __builtin_prefetch(Kh + (size_t)(keyBase + 16) * DD, 0, 1);
      v8f acc = vzero8();
#pragma unroll
      for (int t = 0; t < 16; ++t) {
        const float* bp = Kh + (size_t)(keyBase + ln) * DD + 4 * t + 2 * half;
        v2f bf; bf.x = bp[0]; bf.y = bp[1];
        acc = wmma4(qa[t], bf, acc);
      }
#pragma unroll
      for (int r = 0; r < 8; ++r) {
        int m = r + 8 * half;
        int i = qrow0 + m;
        int tok = keyBase + ln;
        int col = sb * 64 + kt * 16 + ln;
        float x = acc[r] * SCALE;
        myP[m * PSTRIDE + col] = (tok <= i) ? x : NEGINF;
      }
    }
  }

  // exact per-row 256th-largest threshold (radix select on float keys),
  // then thresholded softmax; wave-cooperative, row by row
  for (int r = 0; r < 16; ++r) {
    float* row = myP + r * PSTRIDE;
    unsigned prefix = 0u;
    for (int bit = 31; bit >= 0; --bit) {
      unsigned cand = prefix | (1u << bit);
      int cnt = 0;
      for (int c = lane; c < SELC; c += 32)
        cnt += (okey(row[c]) >= cand) ? 1 : 0;
      cnt = wred_sumi(cnt);
      if (cnt >= KEEP) prefix = cand;  // prefix ends as 256th-largest key
    }
    float mx = -3.4e38f;
    for (int c = lane; c < SELC; c += 32) mx = fmaxf(mx, row[c]);
    mx = wred_maxf(mx);
    float sum = 0.f;
    for (int c = lane; c < SELC; c += 32) {
      float x = row[c];
      float p = (okey(x) >= prefix && mx > -1e29f) ? __expf(x - mx) : 0.f;
      row[c] = p;
      sum += p;
    }
    sum = wred_sumf(sum);
    float rs = (sum > 0.f) ? 1.f / sum : 0.f;
    for (int c = lane; c < SELC; c += 32) row[c] *= rs;
  }

  // out_sel: P(16x512) x gathered V(512x64)
#pragma unroll
  for (int dt = 0; dt < 4; ++dt) {
    for (int g = 0; g < SELC / 4; ++g) {
      int c0 = 4 * g + 2 * half;  // c0, c0+1 always in the same selected block
      v2f a; a.x = myP[ln * PSTRIDE + c0]; a.y = myP[ln * PSTRIDE + c0 + 1];
      int t0 = sSel[c0 >> 6] * BS + (c0 & 63);
      v2f bf;
      bf.x = Vh[(size_t)t0 * DD + dt * 16 + ln];
      bf.y = Vh[(size_t)(t0 + 1) * DD + dt * 16 + ln];
      oacc[dt] = wmma4(a, bf, oacc[dt]);
    }
  }

  // =============== branch 3: sliding window + global token ===============
  const int firstb = (qb >= 2) ? qb - 2 : 0;  // window spans blocks firstb..qb
  const int nwb = qb - firstb + 1;            // 1..3
  const int extra = (firstb > 0) ? 1 : 0;     // prepend block 0 (global token)
  const int ncg = nwb + extra;                // 64-wide column groups, <=4
  const int Cw = ncg * 64;

  for (int cg = 0; cg < ncg; ++cg) {
    int kb = (extra && cg == 0) ? 0 : firstb + (cg - extra);
    for (int kt = 0; kt < 4; ++kt) {
      int keyBase = kb * BS + kt * 16;
      v8f acc = vzero8();
#pragma unroll
      for (int t = 0; t < 16; ++t) {
        const float* bp = Kh + (size_t)(keyBase + ln) * DD + 4 * t + 2 * half;
        v2f bf; bf.x = bp[0]; bf.y = bp[1];
        acc = wmma4(qa[t], bf, acc);
      }
#pragma unroll
      for (int r = 0; r < 8; ++r) {
        int m = r + 8 * half;
        int i = qrow0 + m;
        int j = keyBase + ln;
        int col = cg * 64 + kt * 16 + ln;
        bool ok = (j <= i) && (((i - j) < WIN) || (j < 1));
        myP[m * PSTRIDE + col] = ok ? acc[r] * SCALE : NEGINF;
      }
    }
  }

  for (int r = 0; r < 16; ++r) {
    float* row = myP + r * PSTRIDE;
    float mx = -3.4e38f;
    for (int c = lane; c < Cw; c += 32) mx = fmaxf(mx, row[c]);
    mx = wred_maxf(mx);
    float sum = 0.f;
    for (int c = lane; c < Cw; c += 32) {
      float p = (mx > -1e29f) ? __expf(row[c] - mx) : 0.f;
      row[c] = p;
      sum += p;
    }
    sum = wred_sumf(sum);
    float rs = (sum > 0.f) ? 1.f / sum : 0.f;
    for (int c = lane; c < Cw; c += 32) row[c] *= rs;
  }

  // out_win: P(16xCw) x V(Cwx64)
#pragma unroll
  for (int dt = 0; dt < 4; ++dt) {
    for (int g = 0; g < Cw / 4; ++g) {
      int c0 = 4 * g + 2 * half;
      v2f a; a.x = myP[ln * PSTRIDE + c0]; a.y = myP[ln * PSTRIDE + c0 + 1];
      int cg0 = c0 >> 6;
      int kb0 = (extra && cg0 == 0) ? 0 : firstb + (cg0 - extra);
      int t0 = kb0 * BS + (c0 & 63);
      v2f bf;
      bf.x = Vh[(size_t)t0 * DD + dt * 16 + ln];
      bf.y = Vh[(size_t)(t0 + 1) * DD + dt * 16 + ln];
      oacc[dt] = wmma4(a, bf, oacc[dt]);
    }
  }

  // =============== write (cmp + sel + win) / 3 ===============
  const float third = 1.f / 3.f;
#pragma unroll
  for (int dt = 0; dt < 4; ++dt) {
#pragma unroll
    for (int r = 0; r < 8; ++r) {
      int row = qrow0 + r + 8 * half;  // C-layout: M = r + 8*half, N = ln
      Oh[(size_t)row * DD + dt * 16 + ln] = oacc[dt][r] * third;
    }
  }
}

extern "C" void kernel_launch(void* const* d_in, const int* in_sizes, int n_in,
                              void* d_out, int out_size, void* d_ws,
                              size_t ws_size, hipStream_t stream) {
  (void)in_sizes; (void)n_in; (void)out_size; (void)ws_size;
  const float* q = (const float*)d_in[0];
  const float* k = (const float*)d_in[1];
  const float* v = (const float*)d_in[2];
  float* out = (float*)d_out;
  float* kblk = (float*)d_ws;                       // [B*H*NB, 64]
  float* vblk = kblk + (size_t)BB * HH * NB * DD;   // [B*H*NB, 64]

  nsa_blockmean_kernel<<<BB * HH * NB, 64, 0, stream>>>(k, v, kblk, vblk);

  const size_t shmem = (size_t)64 * PSTRIDE * sizeof(float);  // ~130 KB of 320 KB/WGP
  hipFuncSetAttribute(reinterpret_cast<const void*>(nsa_main_kernel),
                      hipFuncAttributeMaxDynamicSharedMemorySize, (int)shmem);
  nsa_main_kernel<<<BB * HH * NB, 128, shmem, stream>>>(q, k, v, kblk, vblk,
                                                        out);
}